// FRNN_67164698575381
// MI455X (gfx1250) — compile-verified
//
#include <hip/hip_runtime.h>
#include <math.h>

// FRNN persistent-kernel scan for MI455X (gfx1250, wave32).
//   T=8192 steps, C=1024 visible, F=2048 full state.
//   Per step: u = Lam*(W r + b) + (1-Lam)*[x;0];  r' = tanh(u);  out_t = u[:C].
// Strategy: keep all 16 MB of W resident in distributed LDS (64 WGs x 256 KB),
// row-partitioned so no cross-WGP reduction is needed; one grid barrier per
// step; matvec via V_WMMA_F32_16X16X4_F32 with broadcast-B trick.
// This revision: inner K-loop uses loop-carried byte pointers so LDS address
// math folds into ds_load immediate offsets (A stride 256 B, B stride 16 B).

#define T_STEPS 8192
#define C_VIS   1024
#define F_FULL  2048
#define NWG     64          // each WG owns 32 rows of W (2 waves x 16 rows)
#define BLOCK   64          // 2 wave32s
#define KTILES  (F_FULL / 4)   // 512 K-steps of 4 for f32 WMMA

typedef __attribute__((ext_vector_type(2))) float v2f;
typedef __attribute__((ext_vector_type(4))) float v4f;
typedef __attribute__((ext_vector_type(8))) float v8f;

__device__ __forceinline__ v8f wmma_f32x4(v2f a, v2f b, v8f c) {
  // 8-arg pattern: (neg_a, A, neg_b, B, c_mod, C, reuse_a, reuse_b)
  return __builtin_amdgcn_wmma_f32_16x16x4_f32(false, a, false, b,
                                               (short)0, c, false, false);
}

__device__ __forceinline__ void grid_barrier(int* cnt, int* gen, int target) {
  __threadfence();
  if (threadIdx.x == 0) {
    int prev = __hip_atomic_fetch_add(cnt, 1, __ATOMIC_ACQ_REL,
                                      __HIP_MEMORY_SCOPE_AGENT);
    if (prev == NWG - 1) {
      __hip_atomic_store(cnt, 0, __ATOMIC_RELAXED, __HIP_MEMORY_SCOPE_AGENT);
      __hip_atomic_fetch_add(gen, 1, __ATOMIC_RELEASE,
                             __HIP_MEMORY_SCOPE_AGENT);
    } else {
      while (__hip_atomic_load(gen, __ATOMIC_ACQUIRE,
                               __HIP_MEMORY_SCOPE_AGENT) < target) {
        __builtin_amdgcn_s_sleep(2);
      }
    }
  }
  __syncthreads();
}

extern "C" __global__ void frnn_persistent(
    const float* __restrict__ X,    // [T, C]
    const float* __restrict__ W,    // [F, F] row-major
    const float* __restrict__ b,    // [F]
    const float* __restrict__ Lam,  // [F]
    float* __restrict__ out,        // [T, C]
    float* __restrict__ rbuf,       // [2, F] ping-pong state (pre-zeroed)
    int* cnt, int* gen)             // barrier state (pre-zeroed)
{
  extern __shared__ char smem[];
  const int tid  = threadIdx.x;
  const int wave = tid >> 5;
  const int lane = tid & 31;
  const int wg   = blockIdx.x;
  const int row0 = wg * 32 + wave * 16;     // first of this wave's 16 rows
  const int koff = (lane >> 4) << 1;        // 0 for lanes 0-15, 2 for 16-31

  // Per-wave swizzled A region: 512 ktiles * 32 lanes * 8B = 128 KB.
  char*  smemA  = smem + (size_t)wave * (KTILES * 32 * 8);
  v2f*   Aswz   = (v2f*)smemA;
  char*  smemR  = smem + (size_t)2 * (KTILES * 32 * 8);      // 8 KB
  float* rstage = (float*)smemR;

  // ---- one-time: pull this wave's 16 W rows into LDS in WMMA-A layout ----
  // ISA A layout (32-bit 16x4): lane l holds A[M = l&15][K = koff, koff+1].
  // LDS slot (kt,l) -> byte kt*256 + l*8  =>  banks 2l..2l+1: conflict-free.
  {
    const int gr = row0 + (lane & 15);
    const float* wrow = W + (size_t)gr * F_FULL + koff;
    for (int kt = 0; kt < KTILES; ++kt) {
      Aswz[kt * 32 + lane] = *(const v2f*)(wrow + kt * 4);
    }
  }

  // Bias / Lambda for the 8 rows each output lane (0 and 16) will own.
  const int outlane = ((lane & 15) == 0);
  const int rbase   = row0 + ((lane >> 4) << 3);
  float bb[8], ll[8];
  if (outlane) {
#pragma unroll
    for (int j = 0; j < 8; ++j) { bb[j] = b[rbase + j]; ll[j] = Lam[rbase + j]; }
  }

  for (int t = 0; t < T_STEPS; ++t) {
    // ---- stage r_t (8 KB) into LDS, coalesced float4 ----
    const v4f* rc = (const v4f*)(rbuf + (size_t)(t & 1) * F_FULL);
    v4f*       rs = (v4f*)rstage;
#pragma unroll
    for (int j = 0; j < 8; ++j) rs[j * BLOCK + tid] = rc[j * BLOCK + tid];
    __syncthreads();

    // ---- u_partial = W_rows @ r via broadcast-B f32 WMMA ----
    // Loop-carried byte pointers: A advances 256 B/ktile, B 16 B/ktile.
    // Unroll 8 => ds_load immediate offsets 0..1792 (A), 0..112 (B).
    // 4 independent accumulator chains hide WMMA->WMMA RAW latency.
    v8f acc0 = {}, acc1 = {}, acc2 = {}, acc3 = {};
    const char* pA = smemA + lane * 8;
    const char* pB = smemR + koff * 4;
    for (int kt = 0; kt < KTILES; kt += 8) {
      v2f a0 = *(const v2f*)(pA + 0 * 256);
      v2f a1 = *(const v2f*)(pA + 1 * 256);
      v2f a2 = *(const v2f*)(pA + 2 * 256);
      v2f a3 = *(const v2f*)(pA + 3 * 256);
      v2f a4 = *(const v2f*)(pA + 4 * 256);
      v2f a5 = *(const v2f*)(pA + 5 * 256);
      v2f a6 = *(const v2f*)(pA + 6 * 256);
      v2f a7 = *(const v2f*)(pA + 7 * 256);
      // B[k][n] = r[k0+k] for all n (broadcast): lane reads its K-pair.
      v2f b0 = *(const v2f*)(pB + 0 * 16);
      v2f b1 = *(const v2f*)(pB + 1 * 16);
      v2f b2 = *(const v2f*)(pB + 2 * 16);
      v2f b3 = *(const v2f*)(pB + 3 * 16);
      v2f b4 = *(const v2f*)(pB + 4 * 16);
      v2f b5 = *(const v2f*)(pB + 5 * 16);
      v2f b6 = *(const v2f*)(pB + 6 * 16);
      v2f b7 = *(const v2f*)(pB + 7 * 16);
      acc0 = wmma_f32x4(a0, b0, acc0);
      acc1 = wmma_f32x4(a1, b1, acc1);
      acc2 = wmma_f32x4(a2, b2, acc2);
      acc3 = wmma_f32x4(a3, b3, acc3);
      acc0 = wmma_f32x4(a4, b4, acc0);
      acc1 = wmma_f32x4(a5, b5, acc1);
      acc2 = wmma_f32x4(a6, b6, acc2);
      acc3 = wmma_f32x4(a7, b7, acc3);
      pA += 8 * 256;
      pB += 8 * 16;
    }
    v8f acc = (acc0 + acc1) + (acc2 + acc3);

    // ---- epilogue: column N=0 of D lives in lanes 0 (rows 0-7 in v0..v7)
    // and 16 (rows 8-15).  u = Lam*(Wr+b) + (1-Lam)*i;  r' = tanh(u). ----
    float* rn = rbuf + (size_t)((t + 1) & 1) * F_FULL;
    if (outlane) {
#pragma unroll
      for (int j = 0; j < 8; ++j) {
        const int row = rbase + j;
        float u = ll[j] * (acc[j] + bb[j]);
        if (row < C_VIS) {
          u += (1.0f - ll[j]) * X[(size_t)t * C_VIS + row];
          out[(size_t)t * C_VIS + row] = u;
        }
        rn[row] = tanhf(u);
      }
    }

    // ---- all WGPs must see r_{t+1} before next step ----
    grid_barrier(cnt, gen, t + 1);
  }
}

extern "C" void kernel_launch(void* const* d_in, const int* in_sizes, int n_in,
                              void* d_out, int out_size, void* d_ws,
                              size_t ws_size, hipStream_t stream) {
  const float* X   = (const float*)d_in[0];
  const float* W   = (const float*)d_in[1];
  const float* b   = (const float*)d_in[2];
  const float* Lam = (const float*)d_in[3];
  float* out  = (float*)d_out;

  float* rbuf = (float*)d_ws;                                   // 2*F floats
  int*   cnt  = (int*)((char*)d_ws + 2 * F_FULL * sizeof(float));
  int*   gen  = cnt + 1;

  // Zero ping-pong state (r0 = 0) and barrier counters every call so the
  // graph replay is deterministic.
  hipMemsetAsync(d_ws, 0, 2 * F_FULL * sizeof(float) + 2 * sizeof(int),
                 stream);

  const size_t shmem = (size_t)2 * (KTILES * 32 * 8)   // 256 KB W (2 waves)
                     + (size_t)F_FULL * sizeof(float); // 8 KB r stage
  hipFuncSetAttribute((const void*)frnn_persistent,
                      hipFuncAttributeMaxDynamicSharedMemorySize, (int)shmem);

  frnn_persistent<<<NWG, BLOCK, shmem, stream>>>(X, W, b, Lam, out, rbuf,
                                                 cnt, gen);
}